// LatNet_86878598463530
// MI455X (gfx1250) — compile-verified
//
#include <hip/hip_runtime.h>
#include <hip/hip_bf16.h>

// ---------------------------------------------------------------------------
// Problem constants (match reference)
// ---------------------------------------------------------------------------
#define N_NODES 50000
#define HID     128
#define NEDGE   1600000
#define COUT    32
#define EPS_BN  1e-5f
#define FC0_CHUNK 512   // rows of fW0 per block in the big matvec

typedef float v2f __attribute__((ext_vector_type(2)));
typedef float v8f __attribute__((ext_vector_type(8)));

__device__ __forceinline__ float lrelu(float x) { return x > 0.f ? x : 0.1f * x; }

// ---------------------------------------------------------------------------
// Utility: zero a float-addressable region (also used to clear f64 buffers,
// since 0.0f bit pattern == 0.0 bit pattern).
// ---------------------------------------------------------------------------
__global__ void k_zero_f32(float* __restrict__ p, int n) {
  int i = blockIdx.x * blockDim.x + threadIdx.x;
  if (i < n) p[i] = 0.f;
}

// ---------------------------------------------------------------------------
// Degree: deg[col[e]] += 1 ; then dinv = rsqrt(deg + 2)
// ---------------------------------------------------------------------------
__global__ void k_deg_count(const int* __restrict__ col, float* __restrict__ deg) {
  int e = blockIdx.x * blockDim.x + threadIdx.x;
  if (e < NEDGE) atomicAdd(&deg[col[e]], 1.0f);
}

__global__ void k_dinv(const float* __restrict__ deg, float* __restrict__ dinv) {
  int i = blockIdx.x * blockDim.x + threadIdx.x;
  if (i < N_NODES) dinv[i] = rsqrtf(deg[i] + 2.0f);
}

// ---------------------------------------------------------------------------
// XW = X[N,cin] @ W[cin,32] with V_WMMA_F32_16X16X4_F32, one 16x32 strip per
// wave (two accumulators share the A fragment -> two independent WMMAs per
// K-step, hiding the WMMA->WMMA RAW hazard; X is read exactly once).
// Fused epilogue writes both XW and AGG = 2*dinv^2*xw + b (self-loop + bias),
// removing a separate 12.8MB read/write pass per layer.
// N_NODES == 3125*16, COUT == 32 -> no boundary masking, EXEC stays all-1.
// A layout (32-bit A 16x4): lanes 0-15 hold K=k0..k0+1, lanes 16-31 K=k0+2..k0+3.
// ---------------------------------------------------------------------------
__global__ void k_gemm_wmma(const float* __restrict__ X, const float* __restrict__ W,
                            float* __restrict__ XW, float* __restrict__ AGG,
                            const float* __restrict__ dinv, const float* __restrict__ cb,
                            int cin) {
  const int TILES = N_NODES / 16;                          // 3125 strips
  int wave = (blockIdx.x * blockDim.x + threadIdx.x) >> 5; // wave-uniform
  if (wave >= TILES) return;                                // uniform exit
  int lane = threadIdx.x & 31;
  int m0 = wave * 16;
  int l  = lane & 15;
  int hi = lane >> 4;                                       // 0: K lo pair, 1: K hi pair

  const float* xrow = X + (size_t)(m0 + l) * cin + 2 * hi;  // 8B-aligned
  float cb0 = cb[l], cb1 = cb[16 + l];

  v8f acc0 = {}, acc1 = {};
  for (int k0 = 0; k0 < cin; k0 += 4) {
    v2f a = *reinterpret_cast<const v2f*>(xrow + k0);       // A[m0+l][k0+2hi .. +1]
    const float* wr = W + (size_t)(k0 + 2 * hi) * COUT + l; // row k0+2hi of W
    v2f b0, b1;
    b0.x = wr[0];         b0.y = wr[COUT];                  // cols l, rows ka..ka+1
    b1.x = wr[16];        b1.y = wr[COUT + 16];             // cols 16+l
    acc0 = __builtin_amdgcn_wmma_f32_16x16x4_f32(
        false, a, false, b0, (short)0, acc0, false, false);
    acc1 = __builtin_amdgcn_wmma_f32_16x16x4_f32(
        false, a, false, b1, (short)0, acc1, false, false);
  }
  // D layout: VGPR r -> M = m0 + r + 8*hi, N = l (acc0) / 16+l (acc1)
  int mbase = m0 + 8 * hi;
#pragma unroll
  for (int r = 0; r < 8; ++r) {
    int m = mbase + r;
    float di = dinv[m];
    float sl = 2.0f * di * di;
    size_t o = (size_t)m * COUT + l;
    float v0 = acc0[r], v1 = acc1[r];
    XW[o]       = v0;
    XW[o + 16]  = v1;
    AGG[o]      = sl * v0 + cb0;
    AGG[o + 16] = sl * v1 + cb1;
  }
}

// ---------------------------------------------------------------------------
// Edge scatter: one wave per edge, one lane per channel.
// row/col/dinv loads are edge-uniform (scalar broadcast); the 32 atomics per
// edge hit 32 distinct consecutive addresses -> conflict-free within the wave.
// ---------------------------------------------------------------------------
__global__ void k_edge_scatter(const int* __restrict__ row, const int* __restrict__ col,
                               const float* __restrict__ dinv, const float* __restrict__ xw,
                               float* __restrict__ agg) {
  int e = (blockIdx.x * blockDim.x + threadIdx.x) >> 5;
  int c = threadIdx.x & 31;
  if (e >= NEDGE) return;
  int r = row[e], t = col[e];
  float w = dinv[r] * dinv[t];
  atomicAdd(&agg[(size_t)t * COUT + c], xw[(size_t)r * COUT + c] * w);
}

// ---------------------------------------------------------------------------
// In-place LeakyReLU + per-channel sum / sumsq in f64 (LDS reduce -> global).
// Grid covers exactly N*32 elements (1.6M / 256 = 6250 blocks).
// ---------------------------------------------------------------------------
__global__ void k_lrelu_stats(float* __restrict__ y, double* __restrict__ stats) {
  __shared__ double ssum[COUT], ssq[COUT];
  int t = threadIdx.x;
  if (t < COUT) { ssum[t] = 0.0; ssq[t] = 0.0; }
  __syncthreads();
  int i = blockIdx.x * blockDim.x + t;
  float v = lrelu(y[i]);
  y[i] = v;
  int c = i & 31;
  atomicAdd(&ssum[c], (double)v);
  atomicAdd(&ssq[c], (double)v * (double)v);
  __syncthreads();
  if (t < COUT) {
    atomicAdd(&stats[t], ssum[t]);
    atomicAdd(&stats[COUT + t], ssq[t]);
  }
}

// ---------------------------------------------------------------------------
// BatchNorm apply: xo = (y - mean) * rsqrt(var + eps) * g + bt
// ---------------------------------------------------------------------------
__global__ void k_bn_apply(const float* __restrict__ y, const double* __restrict__ stats,
                           const float* __restrict__ g, const float* __restrict__ bt,
                           float* __restrict__ xo) {
  int i = blockIdx.x * blockDim.x + threadIdx.x;
  if (i >= N_NODES * COUT) return;
  int c = i & 31;
  double mean = stats[c] * (1.0 / N_NODES);
  double var  = stats[COUT + c] * (1.0 / N_NODES) - mean * mean;
  xo[i] = (y[i] - (float)mean) * rsqrtf((float)var + EPS_BN) * g[c] + bt[c];
}

// ---------------------------------------------------------------------------
// FC0: acc[j] += sum_i x[i] * fW0[i*128 + j], 512 rows per block (128 threads,
// thread j owns column j -> fully coalesced 512B rows of fW0). 819MB stream,
// prefetched 32 rows (16KB) ahead to stay at line rate.
// ---------------------------------------------------------------------------
__global__ void k_fc0_partial(const float* __restrict__ x, const float* __restrict__ w,
                              double* __restrict__ acc) {
  int j = threadIdx.x;  // 0..127
  size_t base = (size_t)blockIdx.x * FC0_CHUNK;
  const float* wp = w + base * 128 + j;
  float s = 0.f;
  for (int i = 0; i < FC0_CHUNK; ++i) {
    if (i + 32 < FC0_CHUNK)
      __builtin_prefetch(wp + (size_t)(i + 32) * 128, 0, 0);  // global_prefetch_b8
    s += x[base + i] * wp[(size_t)i * 128];
  }
  atomicAdd(&acc[j], (double)s);
}

// ---------------------------------------------------------------------------
// Tiny MLP head: 128 -> 64 -> 32 -> 1, single block of 128 threads.
// ---------------------------------------------------------------------------
__global__ void k_fc_head(const double* __restrict__ acc0, const float* __restrict__ fb0,
                          const float* __restrict__ fW1, const float* __restrict__ fb1,
                          const float* __restrict__ fW2, const float* __restrict__ fb2,
                          const float* __restrict__ fW3, const float* __restrict__ fb3,
                          float* __restrict__ out) {
  __shared__ float h0[128], h1[64], h2[32];
  int t = threadIdx.x;
  h0[t] = lrelu((float)acc0[t] + fb0[t]);
  __syncthreads();
  if (t < 64) {
    float s = fb1[t];
    for (int j = 0; j < 128; ++j) s += h0[j] * fW1[j * 64 + t];
    h1[t] = lrelu(s);
  }
  __syncthreads();
  if (t < 32) {
    float s = fb2[t];
    for (int j = 0; j < 64; ++j) s += h1[j] * fW2[j * 32 + t];
    h2[t] = lrelu(s);
  }
  __syncthreads();
  if (t == 0) {
    float s = fb3[0];
    for (int j = 0; j < 32; ++j) s += h2[j] * fW3[j];
    out[0] = s;
  }
}

// ---------------------------------------------------------------------------
// Orchestration
// ---------------------------------------------------------------------------
extern "C" void kernel_launch(void* const* d_in, const int* in_sizes, int n_in,
                              void* d_out, int out_size, void* d_ws, size_t ws_size,
                              hipStream_t stream) {
  const int*   edge = (const int*)d_in[0];
  const int*   erow = edge;          // edge_index[0,:]
  const int*   ecol = edge + NEDGE;  // edge_index[1,:]
  const float* emb  = (const float*)d_in[1];
  const float* cW[3] = {(const float*)d_in[2], (const float*)d_in[6],  (const float*)d_in[10]};
  const float* cb[3] = {(const float*)d_in[3], (const float*)d_in[7],  (const float*)d_in[11]};
  const float* g [3] = {(const float*)d_in[4], (const float*)d_in[8],  (const float*)d_in[12]};
  const float* bt[3] = {(const float*)d_in[5], (const float*)d_in[9],  (const float*)d_in[13]};
  const float* fW0 = (const float*)d_in[14]; const float* fb0 = (const float*)d_in[15];
  const float* fW1 = (const float*)d_in[16]; const float* fb1 = (const float*)d_in[17];
  const float* fW2 = (const float*)d_in[18]; const float* fb2 = (const float*)d_in[19];
  const float* fW3 = (const float*)d_in[20]; const float* fb3 = (const float*)d_in[21];

  // Workspace carve (needs ~14.5 MB): two ping-pong node buffers + small state.
  char*   ws    = (char*)d_ws;
  float*  deg   = (float*)(ws + 0x000000);   // 200 KB
  float*  dinv  = (float*)(ws + 0x040000);   // 200 KB
  double* stats = (double*)(ws + 0x080000);  // 64 f64
  double* acc0  = (double*)(ws + 0x081000);  // 128 f64
  float*  bufA  = (float*)(ws + 0x100000);   // 6.4 MB
  float*  bufB  = (float*)(ws + 0x800000);   // 6.4 MB

  const int NC = N_NODES * COUT;             // 1,600,000
  const int TILES = N_NODES / 16;            // 3125 WMMA strips (16x32 each)

  // --- degrees / normalization ---
  k_zero_f32 <<<(N_NODES + 255) / 256, 256, 0, stream>>>(deg, N_NODES);
  k_deg_count<<<(NEDGE   + 255) / 256, 256, 0, stream>>>(ecol, deg);
  k_dinv     <<<(N_NODES + 255) / 256, 256, 0, stream>>>(deg, dinv);

  // --- 3 GCN layers, ping-pong between bufA/bufB ---
  struct Layer { const float* X; float* XW; float* AGG; float* OUT; int cin; };
  Layer L[3] = {
    { emb,  bufA, bufB, bufA, HID  },   // L0: xw->A, agg->B, bn B->A
    { bufA, bufB, bufA, bufB, COUT },   // L1: xw->B, agg->A, bn A->B
    { bufB, bufA, bufB, bufA, COUT },   // L2: xw->A, agg->B, bn B->A
  };
  for (int l = 0; l < 3; ++l) {
    k_gemm_wmma   <<<(TILES * 32 + 255) / 256, 256, 0, stream>>>(
        L[l].X, cW[l], L[l].XW, L[l].AGG, dinv, cb[l], L[l].cin);
    k_edge_scatter<<<(NEDGE * 32 + 255)/256,  256, 0, stream>>>(erow, ecol, dinv, L[l].XW, L[l].AGG);
    k_zero_f32    <<<1, 128, 0, stream>>>((float*)stats, 128);  // 64 f64 == 128 f32 slots
    k_lrelu_stats <<<NC / 256,                256, 0, stream>>>(L[l].AGG, stats);
    k_bn_apply    <<<(NC + 255) / 256,        256, 0, stream>>>(L[l].AGG, stats, g[l], bt[l], L[l].OUT);
  }

  // --- MLP head: big matvec then tiny chain ---
  k_zero_f32    <<<1, 256, 0, stream>>>((float*)acc0, 256);     // 128 f64
  k_fc0_partial <<<NC / FC0_CHUNK, 128, 0, stream>>>(bufA, fW0, acc0);
  k_fc_head     <<<1, 128, 0, stream>>>(acc0, fb0, fW1, fb1, fW2, fb2, fW3, fb3, (float*)d_out);
}